// Navigator_50148038148775
// MI455X (gfx1250) — compile-verified
//
#include <hip/hip_runtime.h>
#include <math.h>

typedef __attribute__((ext_vector_type(2))) float v2f;
typedef __attribute__((ext_vector_type(8))) float v8f;
typedef unsigned int u32x4 __attribute__((ext_vector_type(4)));
typedef int i32x4 __attribute__((ext_vector_type(4)));
typedef int i32x8 __attribute__((ext_vector_type(8)));

#define L_SEQ   100
#define NBATCH  256
#define M_ROWS  (NBATCH * L_SEQ)   // 25600
#define KDIM    3200
#define FC      64
#define HDIM    256
#define G4      1024
#define RNN_IN  128
#define KCH     128                 // K-chunk of W_cnn staged in LDS (32KB)
#define NCH     (KDIM / KCH)        // 25 chunks

#if __has_builtin(__builtin_amdgcn_tensor_load_to_lds) && \
    __has_builtin(__builtin_amdgcn_s_wait_tensorcnt)
#define HAVE_TDM 1
#else
#define HAVE_TDM 0
#endif

// fp32 WMMA: D(16x16) = A(16x4) * B(4x16) + C.  8-arg form:
// (neg_a, A, neg_b, B, c_mod, C, reuse_a, reuse_b)
__device__ __forceinline__ v8f wmma4(v2f a, v2f b, v8f c) {
    return __builtin_amdgcn_wmma_f32_16x16x4_f32(false, a, false, b,
                                                 (short)0, c, false, false);
}

__device__ __forceinline__ v2f ld2(const float* p) {
    float2 t = *(const float2*)p;
    v2f a; a.x = t.x; a.y = t.y;
    return a;
}

__device__ __forceinline__ float sigmoidf_(float x) {
    return 1.0f / (1.0f + expf(-x));
}

#if HAVE_TDM
// TDM: DMA a contiguous [rows x 64] f32 tile (row stride 64) from global
// memory into LDS at lds_byte_off.  D# per CDNA5 ISA ch.10.7/8.
__device__ __forceinline__ void tdm_load_rows64(const void* gaddr,
                                                unsigned lds_byte_off,
                                                unsigned rows) {
    unsigned long long ga = (unsigned long long)(uintptr_t)gaddr;
    u32x4 g0 = {
        1u,                                        // count=1, user descriptor
        lds_byte_off,                              // LDS byte address
        (unsigned)(ga & 0xffffffffu),              // global_addr[31:0]
        (unsigned)((ga >> 32) & 0x01ffffffu) | (2u << 30)  // addr[56:32], type=2
    };
    i32x8 g1 = {
        (int)(2u << 16),                 // data_size = 2 (4 bytes)
        (int)(64u << 16),                // tensor_dim0 = 64 (bits 79:48)
        (int)((rows & 0xffffu) << 16),   // tensor_dim1[15:0] (bits 111:80)
        (int)(64u << 16),                // tile_dim0 = 64 (bits 127:112)
        (int)(rows & 0xffffu),           // tile_dim1 = rows (bits 143:128)
        64,                              // tensor_dim0_stride = 64 (bits 207:160)
        0, 0
    };
    i32x4 gz = {0, 0, 0, 0};
#if __clang_major__ >= 23
    i32x8 gz8 = {0, 0, 0, 0, 0, 0, 0, 0};
    __builtin_amdgcn_tensor_load_to_lds(g0, g1, gz, gz, gz8, 0);
#else
    __builtin_amdgcn_tensor_load_to_lds(g0, g1, gz, gz, 0);
#endif
}
#endif

// ---------------- init: zero c-state and grid-barrier counter ----------------
__global__ void init_kernel(float* cbuf, unsigned* cnt) {
    int idx = blockIdx.x * blockDim.x + threadIdx.x;
    if (idx < NBATCH * HDIM) cbuf[idx] = 0.0f;
    if (idx == 0) *cnt = 0u;
}

// -------- pack B matrices into k-paired layout for 8-byte WMMA fragments -----
// P[(k>>1)*2N + n*2 + (k&1)] = W[k*N + n]  so (B[k][n], B[k+1][n]) is one b64.
__global__ void pack_kernel(const float* __restrict__ Wc,
                            const float* __restrict__ Wih,
                            const float* __restrict__ Wa1,
                            float* __restrict__ Wcp,
                            float* __restrict__ Wihp,
                            float* __restrict__ Wa1p) {
    int idx = blockIdx.x * blockDim.x + threadIdx.x;
    if (idx < KDIM * FC) {
        int k = idx >> 6, n = idx & 63;
        Wcp[(k >> 1) * 128 + n * 2 + (k & 1)] = Wc[idx];
    } else if (idx < KDIM * FC + RNN_IN * G4) {
        int j = idx - KDIM * FC;
        int k = j >> 10, n = j & 1023;
        Wihp[(k >> 1) * 2048 + n * 2 + (k & 1)] = Wih[j];
    } else if (idx < KDIM * FC + RNN_IN * G4 + HDIM * FC) {
        int j = idx - KDIM * FC - RNN_IN * G4;
        int k = j >> 6, n = j & 63;
        Wa1p[(k >> 1) * 128 + n * 2 + (k & 1)] = Wa1[j];
    }
}

// ---------------- phrase embed: relu(phr @ W_phr + b) -> (256,64) ------------
__global__ void phrase_kernel(const float* __restrict__ phr,
                              const float* __restrict__ W,
                              const float* __restrict__ b,
                              float* __restrict__ out) {
    int idx = blockIdx.x * blockDim.x + threadIdx.x;   // 16384 total
    int n = idx >> 6, f = idx & 63;
    const float* p = phr + (size_t)n * 300;
    float s = b[f];
    for (int k = 0; k < 300; ++k) s += p[k] * W[k * FC + f];
    out[idx] = fmaxf(s, 0.0f);
}

// ---------------- action embedding gather into x[:, 64:128] ------------------
__global__ void act_embed_kernel(const int* __restrict__ acts,
                                 const float* __restrict__ emb,
                                 float* __restrict__ x) {
    int idx = blockIdx.x * blockDim.x + threadIdx.x;
    if (idx >= M_ROWS * FC) return;
    int row = idx >> 6, f = idx & 63;
    int id = acts[row];
    x[(size_t)row * RNN_IN + FC + f] = emb[id * FC + f];
}

// ------- img GEMM: x[:, :64] = relu(img @ W_cnn + b) * emb_fc[sample] --------
// wave = one 16-row M tile, 4 accumulators span all 64 output cols.
// Packed W_cnn is staged through LDS in 32KB chunks by the Tensor Data Mover
// (double-buffered); each B fragment is a single ds_load_b64.
__global__ void img_gemm_kernel(const float* __restrict__ img,
                                const float* __restrict__ Wcp,
                                const float* __restrict__ bc,
                                const float* __restrict__ embfc,
                                float* __restrict__ x) {
    __shared__ float Bsh[2][KCH * FC];                 // 2 x 32KB
    const int tid = threadIdx.x, lane = tid & 31, wid = tid >> 5;
    const int mt = blockIdx.x * 8 + wid;               // 0..1599
    const int ln = lane & 15, half = lane >> 4, kb = half * 2;
    v8f acc[4] = {};
    const float* arow = img + (size_t)(mt * 16 + ln) * KDIM;

#if HAVE_TDM
    if (wid == 0) {
        tdm_load_rows64(Wcp, (unsigned)(uintptr_t)&Bsh[0][0], KCH);
        __builtin_amdgcn_s_wait_tensorcnt(0);
    }
    __syncthreads();
    for (int c = 0; c < NCH; ++c) {
        const int cur = c & 1;
        if (wid == 0 && c + 1 < NCH)
            tdm_load_rows64(Wcp + (size_t)(c + 1) * KCH * FC,
                            (unsigned)(uintptr_t)&Bsh[cur ^ 1][0], KCH);
        const float* bs = &Bsh[cur][0];
        const int cbase = c * KCH;
        for (int k0 = 0; k0 < KCH; k0 += 4) {
            int kA = k0 + kb;
            v2f a = ld2(arow + cbase + kA);
#pragma unroll
            for (int nt = 0; nt < 4; ++nt) {
                v2f b = ld2(bs + (kA >> 1) * 128 + (nt * 16 + ln) * 2);
                acc[nt] = wmma4(a, b, acc[nt]);
            }
        }
        if (wid == 0 && c + 1 < NCH) __builtin_amdgcn_s_wait_tensorcnt(0);
        __syncthreads();
    }
#else
    for (int c = 0; c < NCH; ++c) {
        __syncthreads();
        for (int i = tid; i < KCH * FC; i += 256)
            Bsh[0][i] = Wcp[(size_t)c * KCH * FC + i];
        __syncthreads();
        const float* bs = &Bsh[0][0];
        const int cbase = c * KCH;
        for (int k0 = 0; k0 < KCH; k0 += 4) {
            int kA = k0 + kb;
            v2f a = ld2(arow + cbase + kA);
#pragma unroll
            for (int nt = 0; nt < 4; ++nt) {
                v2f b = ld2(bs + (kA >> 1) * 128 + (nt * 16 + ln) * 2);
                acc[nt] = wmma4(a, b, acc[nt]);
            }
        }
    }
#endif

#pragma unroll
    for (int nt = 0; nt < 4; ++nt) {
        int f = nt * 16 + ln;
        float bias = bc[f];
#pragma unroll
        for (int r = 0; r < 8; ++r) {
            int row = mt * 16 + r + half * 8;
            int sample = row / L_SEQ;
            float v = fmaxf(acc[nt][r] + bias, 0.0f) * embfc[sample * FC + f];
            x[(size_t)row * RNN_IN + f] = v;
        }
    }
}

// ---------------- persistent LSTM: 16 blocks, each owns 16 h-columns ---------
// gates = [x_t ; h_{t-1}] @ [W_ih ; W_hh] + b  (K = 384, fused, no gx tensor)
__global__ void lstm_kernel(const float* __restrict__ x,
                            const float* __restrict__ Wihp,   // packed
                            const float* __restrict__ W_hh,
                            const float* __restrict__ b_ih,
                            const float* __restrict__ b_hh,
                            float* __restrict__ feats,   // (n, L, 256) output region
                            float* __restrict__ hlast,
                            float* __restrict__ clast,
                            float* __restrict__ cbuf,
                            unsigned* cnt) {
    __shared__ float Bsh[HDIM * 64];   // 64KB: W_hh slice, k-paired swizzle
    const int tid = threadIdx.x, lane = tid & 31, wid = tid >> 5;
    const int s0 = blockIdx.x * 16;                    // owned h-column base
    for (int idx = tid; idx < HDIM * 64; idx += 256) {
        int k = idx >> 6, col = idx & 63, g = col >> 4, n = col & 15;
        Bsh[(k >> 1) * 128 + col * 2 + (k & 1)] =
            W_hh[(size_t)k * G4 + g * HDIM + s0 + n];
    }
    __syncthreads();

    const int ln = lane & 15, half = lane >> 4, kb = half * 2;
    float bias[4];
#pragma unroll
    for (int g = 0; g < 4; ++g) {
        int c = g * HDIM + s0 + ln;
        bias[g] = b_ih[c] + b_hh[c];
    }

    for (int t = 0; t < L_SEQ; ++t) {
        for (int mpass = 0; mpass < 2; ++mpass) {
            const int mt = wid + mpass * 8;            // 0..15 M-tiles (batch)
            const int row = mt * 16 + ln;              // batch index for A loads
            v8f acc[4] = {};
            // x contribution (K = 0..127), packed W_ih from L2 (b64 fragments)
            for (int k0 = 0; k0 < RNN_IN; k0 += 4) {
                int kA = k0 + kb;
                v2f a = ld2(x + ((size_t)row * L_SEQ + t) * RNN_IN + kA);
#pragma unroll
                for (int g = 0; g < 4; ++g) {
                    v2f b = ld2(Wihp + (size_t)(kA >> 1) * 2048 +
                                (g * HDIM + s0 + ln) * 2);
                    acc[g] = wmma4(a, b, acc[g]);
                }
            }
            // h contribution (K = 0..255), B from LDS (b64 fragments)
            for (int k0 = 0; k0 < HDIM; k0 += 4) {
                int kA = k0 + kb;
                v2f a;
                if (t == 0) { a.x = 0.0f; a.y = 0.0f; }
                else a = ld2(feats + ((size_t)row * L_SEQ + (t - 1)) * HDIM + kA);
#pragma unroll
                for (int g = 0; g < 4; ++g) {
                    v2f b = ld2(Bsh + (kA >> 1) * 128 + (g * 16 + ln) * 2);
                    acc[g] = wmma4(a, b, acc[g]);
                }
            }
            // elementwise cell update: gate order i, f, g, o
            const int colh = s0 + ln;
#pragma unroll
            for (int r = 0; r < 8; ++r) {
                int rown = mt * 16 + r + half * 8;
                float ig = sigmoidf_(acc[0][r] + bias[0]);
                float fg = sigmoidf_(acc[1][r] + bias[1]);
                float gg = tanhf(acc[2][r] + bias[2]);
                float og = sigmoidf_(acc[3][r] + bias[3]);
                size_t ci = (size_t)rown * HDIM + colh;
                float cnew = fg * cbuf[ci] + ig * gg;
                cbuf[ci] = cnew;
                feats[((size_t)rown * L_SEQ + t) * HDIM + colh] = og * tanhf(cnew);
            }
        }
        // device-wide step barrier (16 resident blocks)
        __syncthreads();
        __threadfence();
        if (tid == 0) {
            atomicAdd(cnt, 1u);
            unsigned target = (unsigned)(t + 1) * (unsigned)gridDim.x;
            while (atomicAdd(cnt, 0u) < target) __builtin_amdgcn_s_sleep(2);
        }
        __syncthreads();
    }
    // each block writes its owned columns of h_last / c_last
    for (int idx = tid; idx < NBATCH * 16; idx += 256) {
        int rown = idx >> 4, coll = s0 + (idx & 15);
        hlast[(size_t)rown * HDIM + coll] =
            feats[((size_t)rown * L_SEQ + (L_SEQ - 1)) * HDIM + coll];
        clast[(size_t)rown * HDIM + coll] = cbuf[(size_t)rown * HDIM + coll];
    }
}

// ---------------- head: relu(feats @ W_a1 + b1) -> logits -> log_softmax -----
__global__ void head_kernel(const float* __restrict__ feats,
                            const float* __restrict__ W1p,   // packed
                            const float* __restrict__ b1,
                            const float* __restrict__ W2,
                            const float* __restrict__ b2,
                            float* __restrict__ logp) {
    __shared__ float hid[32 * 64];
    __shared__ float lg[32 * 4];
    const int tid = threadIdx.x, lane = tid & 31, wid = tid >> 5;
    const int mtl = wid >> 2, nt = wid & 3;
    const int mt = blockIdx.x * 2 + mtl;               // 0..1599
    const int ln = lane & 15, half = lane >> 4, kb = half * 2;
    v8f acc = {};
    const float* arow = feats + (size_t)(mt * 16 + ln) * HDIM;
    for (int k0 = 0; k0 < HDIM; k0 += 4) {
        int kA = k0 + kb;
        v2f a = ld2(arow + kA);
        v2f b = ld2(W1p + (size_t)(kA >> 1) * 128 + (nt * 16 + ln) * 2);
        acc = wmma4(a, b, acc);
    }
    int f = nt * 16 + ln;
#pragma unroll
    for (int r = 0; r < 8; ++r) {
        int lr = mtl * 16 + r + half * 8;
        hid[lr * 64 + f] = fmaxf(acc[r] + b1[f], 0.0f);
    }
    __syncthreads();
    if (tid < 128) {
        int r = tid >> 2, a = tid & 3;
        float s = b2[a];
        for (int k = 0; k < FC; ++k) s += hid[r * 64 + k] * W2[k * 4 + a];
        lg[r * 4 + a] = s;
    }
    __syncthreads();
    if (tid < 32) {
        int r = tid;
        float l0 = lg[r * 4 + 0], l1 = lg[r * 4 + 1];
        float l2 = lg[r * 4 + 2], l3 = lg[r * 4 + 3];
        float m = fmaxf(fmaxf(l0, l1), fmaxf(l2, l3));
        float se = expf(l0 - m) + expf(l1 - m) + expf(l2 - m) + expf(l3 - m);
        float lse = m + logf(se);
        size_t rowg = (size_t)blockIdx.x * 32 + r;
        logp[rowg * 4 + 0] = l0 - lse;
        logp[rowg * 4 + 1] = l1 - lse;
        logp[rowg * 4 + 2] = l2 - lse;
        logp[rowg * 4 + 3] = l3 - lse;
    }
}

extern "C" void kernel_launch(void* const* d_in, const int* in_sizes, int n_in,
                              void* d_out, int out_size, void* d_ws, size_t ws_size,
                              hipStream_t stream) {
    (void)in_sizes; (void)n_in; (void)out_size; (void)ws_size;
    const float* img  = (const float*)d_in[0];
    const float* phr  = (const float*)d_in[1];
    const int*   acts = (const int*)  d_in[2];
    const float* Wc   = (const float*)d_in[3];
    const float* bc   = (const float*)d_in[4];
    const float* Wp   = (const float*)d_in[5];
    const float* bp   = (const float*)d_in[6];
    const float* emb  = (const float*)d_in[7];
    const float* Wih  = (const float*)d_in[8];
    const float* Whh  = (const float*)d_in[9];
    const float* bih  = (const float*)d_in[10];
    const float* bhh  = (const float*)d_in[11];
    const float* Wa1  = (const float*)d_in[12];
    const float* ba1  = (const float*)d_in[13];
    const float* Wa2  = (const float*)d_in[14];
    const float* ba2  = (const float*)d_in[15];

    float* out   = (float*)d_out;
    float* logp  = out;                                   // 256*100*4
    float* feats = out + (size_t)M_ROWS * 4;              // 256*100*256
    float* hlast = feats + (size_t)M_ROWS * HDIM;         // 256*256
    float* clast = hlast + (size_t)NBATCH * HDIM;

    float* ws    = (float*)d_ws;
    float* x     = ws;                                    // 25600*128
    float* embfc = x + (size_t)M_ROWS * RNN_IN;           // 256*64
    float* cbuf  = embfc + (size_t)NBATCH * FC;           // 256*256
    unsigned* cnt = (unsigned*)(cbuf + (size_t)NBATCH * HDIM);
    float* Wcp   = cbuf + (size_t)NBATCH * HDIM + 64;     // 3200*64 packed
    float* Wihp  = Wcp + (size_t)KDIM * FC;               // 128*1024 packed
    float* Wa1p  = Wihp + (size_t)RNN_IN * G4;            // 256*64 packed

    const int pack_total = KDIM * FC + RNN_IN * G4 + HDIM * FC;  // 352256

    hipLaunchKernelGGL(init_kernel, dim3(256), dim3(256), 0, stream, cbuf, cnt);
    hipLaunchKernelGGL(pack_kernel, dim3((pack_total + 255) / 256), dim3(256), 0,
                       stream, Wc, Wih, Wa1, Wcp, Wihp, Wa1p);
    hipLaunchKernelGGL(phrase_kernel, dim3(64), dim3(256), 0, stream, phr, Wp, bp, embfc);
    hipLaunchKernelGGL(act_embed_kernel, dim3((M_ROWS * FC + 255) / 256), dim3(256), 0,
                       stream, acts, emb, x);
    hipLaunchKernelGGL(img_gemm_kernel, dim3(200), dim3(256), 0, stream,
                       img, Wcp, bc, embfc, x);
    hipLaunchKernelGGL(lstm_kernel, dim3(16), dim3(256), 0, stream,
                       x, Wihp, Whh, bih, bhh, feats, hlast, clast, cbuf, cnt);
    hipLaunchKernelGGL(head_kernel, dim3(800), dim3(256), 0, stream,
                       feats, Wa1p, ba1, Wa2, ba2, logp);
}